// STGAT_24043226923663
// MI455X (gfx1250) — compile-verified
//
#include <hip/hip_runtime.h>
#include <math.h>

#define N_NODES 100000
#define N_EDGES 1600000
#define IN_DIM  8
#define HID     64
#define OUT_DIM 32
#define LN_EPS  1e-5f

typedef __attribute__((ext_vector_type(2))) float v2f;
typedef __attribute__((ext_vector_type(8))) float v8f;

// ------------------------------------------------------------------
// WMMA f32 GEMM:  C[N,64] = A[N,K] @ W[K,64]   (K multiple of 4)
// One wave computes a 16x16 tile via V_WMMA_F32_16X16X4_F32.
// Block = 128 threads = 4 waves -> covers all 64 output cols of a
// 16-row stripe; grid = N/16 row stripes.
// A-tile layout (ISA 7.12.2): lanes 0-15 hold M=0..15 K=(k,k+1) in
// VGPR0/1; lanes 16-31 hold K=(k+2,k+3). B mirrors per-lane. C/D:
// VGPR r -> M=r (lanes 0-15) / M=r+8 (lanes 16-31), N=lane%16.
// ------------------------------------------------------------------
__global__ void gemm_wmma_f32(const float* __restrict__ A,
                              const float* __restrict__ W,
                              float* __restrict__ C, int K) {
  const int wave = threadIdx.x >> 5;
  const int lane = threadIdx.x & 31;
  const int hl   = lane >> 4;          // 0: K+0/1   1: K+2/3
  const int l16  = lane & 15;
  const int row0 = blockIdx.x * 16;
  const int col0 = wave * 16;
  v8f acc = {};
  for (int k = 0; k < K; k += 4) {
    v2f a, b;
    const float* ap = A + (size_t)(row0 + l16) * K + (k + 2 * hl);
    a.x = ap[0];
    a.y = ap[1];
    b.x = W[(size_t)(k + 2 * hl)     * HID + col0 + l16];
    b.y = W[(size_t)(k + 2 * hl + 1) * HID + col0 + l16];
    acc = __builtin_amdgcn_wmma_f32_16x16x4_f32(false, a, false, b,
                                                (short)0, acc, false, false);
  }
#pragma unroll
  for (int r = 0; r < 8; ++r)
    C[(size_t)(row0 + 8 * hl + r) * HID + col0 + l16] = acc[r];
}

// ------------------------------------------------------------------
// prep: scal[0] = sum(a[0:128]); scal[1] = -inf (max); scal[2] = 0 (sum)
// ------------------------------------------------------------------
__global__ void prep_scal(const float* __restrict__ a, float* __restrict__ scal) {
  __shared__ float sh[2 * HID];
  sh[threadIdx.x] = a[threadIdx.x];
  __syncthreads();
  for (int s = HID; s > 0; s >>= 1) {
    if ((int)threadIdx.x < s) sh[threadIdx.x] += sh[threadIdx.x + s];
    __syncthreads();
  }
  if (threadIdx.x == 0) {
    scal[0] = sh[0];
    scal[1] = -INFINITY;
    scal[2] = 0.0f;
  }
}

__device__ inline void atomicMaxF(float* addr, float v) {
  unsigned int* ua = (unsigned int*)addr;
  unsigned int old = *ua;
  while (__uint_as_float(old) < v) {
    unsigned int assumed = old;
    old = atomicCAS(ua, assumed, __float_as_uint(v));
    if (old == assumed) break;
  }
}

// ------------------------------------------------------------------
// per-edge score: s = dot(Wh[src],Wh[dst]) * sum(a); leaky_relu(0.2)
// wave32 max-reduce, one float atomicMax per wave.
// ------------------------------------------------------------------
__global__ void edge_scores_k(const float* __restrict__ Wh,
                              const int* __restrict__ src,
                              const int* __restrict__ dst,
                              const float* __restrict__ scal,
                              float* __restrict__ scores,
                              float* __restrict__ gmax) {
  int e = blockIdx.x * blockDim.x + threadIdx.x;
  float v = -INFINITY;
  if (e < N_EDGES) {
    const float4* ps = (const float4*)(Wh + (size_t)src[e] * HID);
    const float4* pd = (const float4*)(Wh + (size_t)dst[e] * HID);
    float s = 0.f;
#pragma unroll
    for (int j = 0; j < HID / 4; ++j) {
      float4 a = ps[j], b = pd[j];
      s += a.x * b.x + a.y * b.y + a.z * b.z + a.w * b.w;
    }
    s *= scal[0];
    v = (s > 0.f) ? s : 0.2f * s;
    scores[e] = v;
  }
  for (int off = 16; off > 0; off >>= 1)
    v = fmaxf(v, __shfl_xor(v, off, 32));
  if ((threadIdx.x & 31) == 0) atomicMaxF(gmax, v);
}

// ------------------------------------------------------------------
// w = exp(score - gmax); global sum via wave reduce + atomicAdd
// ------------------------------------------------------------------
__global__ void edge_expsum_k(float* __restrict__ scores,
                              const float* __restrict__ scal,  // [1] = gmax
                              float* __restrict__ gsum) {
  int e = blockIdx.x * blockDim.x + threadIdx.x;
  float w = 0.f;
  if (e < N_EDGES) {
    w = expf(scores[e] - scal[1]);
    scores[e] = w;
  }
  for (int off = 16; off > 0; off >>= 1)
    w += __shfl_xor(w, off, 32);
  if ((threadIdx.x & 31) == 0) atomicAdd(gsum, w);
}

// ------------------------------------------------------------------
// scatter: hpr[dst[e]][j] += (w_e / gsum) * Wh[src[e]][j]
// one thread per (edge, col) -> GLOBAL_ATOMIC_ADD_F32 (L2-resident)
// ------------------------------------------------------------------
__global__ void scatter_k(const float* __restrict__ scores,
                          const float* __restrict__ scal,  // [2] = gsum
                          const float* __restrict__ Wh,
                          const int* __restrict__ src,
                          const int* __restrict__ dst,
                          float* __restrict__ hpr) {
  long long t = (long long)blockIdx.x * blockDim.x + threadIdx.x;
  if (t >= (long long)N_EDGES * HID) return;
  int e = (int)(t >> 6);
  int j = (int)(t & (HID - 1));
  float att = scores[e] / scal[2];
  atomicAdd(hpr + (size_t)dst[e] * HID + j, att * Wh[(size_t)src[e] * HID + j]);
}

// ------------------------------------------------------------------
// fused ELU + LayerNorm; one wave32 per 64-wide row (2 elems/lane)
// ------------------------------------------------------------------
__global__ void elu_ln_k(const float* __restrict__ hpr,
                         const float* __restrict__ g,
                         const float* __restrict__ b,
                         float* __restrict__ out) {
  int lane = threadIdx.x & 31;
  int node = blockIdx.x * (blockDim.x >> 5) + (threadIdx.x >> 5);
  if (node >= N_NODES) return;
  float x0 = hpr[(size_t)node * HID + lane];
  float x1 = hpr[(size_t)node * HID + lane + 32];
  x0 = (x0 > 0.f) ? x0 : (expf(x0) - 1.f);
  x1 = (x1 > 0.f) ? x1 : (expf(x1) - 1.f);
  float s = x0 + x1;
  for (int off = 16; off > 0; off >>= 1) s += __shfl_xor(s, off, 32);
  float mean = s * (1.0f / HID);
  float d0 = x0 - mean, d1 = x1 - mean;
  float vv = d0 * d0 + d1 * d1;
  for (int off = 16; off > 0; off >>= 1) vv += __shfl_xor(vv, off, 32);
  float inv = rsqrtf(vv * (1.0f / HID) + LN_EPS);
  out[(size_t)node * HID + lane]      = g[lane]      * d0 * inv + b[lane];
  out[(size_t)node * HID + lane + 32] = g[lane + 32] * d1 * inv + b[lane + 32];
}

// ------------------------------------------------------------------
// head: h = sp+tp; relu(h@cW1+cb1) (lane = out col); sigmoid(.@cW2+cb2)
// ------------------------------------------------------------------
__global__ void head_k(const float* __restrict__ sp,
                       const float* __restrict__ tp,
                       const float* __restrict__ cW1,
                       const float* __restrict__ cb1,
                       const float* __restrict__ cW2,
                       const float* __restrict__ cb2,
                       float* __restrict__ out) {
  int lane = threadIdx.x & 31;
  int node = blockIdx.x * (blockDim.x >> 5) + (threadIdx.x >> 5);
  if (node >= N_NODES) return;
  const float* ps = sp + (size_t)node * HID;
  const float* pt = tp + (size_t)node * HID;
  float acc = cb1[lane];
#pragma unroll 8
  for (int k = 0; k < HID; ++k)
    acc += (ps[k] + pt[k]) * cW1[k * OUT_DIM + lane];
  float r = fmaxf(acc, 0.f);
  float z = r * cW2[lane];
  for (int off = 16; off > 0; off >>= 1) z += __shfl_xor(z, off, 32);
  if (lane == 0) out[node] = 1.f / (1.f + expf(-(z + cb2[0])));
}

extern "C" void kernel_launch(void* const* d_in, const int* in_sizes, int n_in,
                              void* d_out, int out_size, void* d_ws, size_t ws_size,
                              hipStream_t stream) {
  const float* x  = (const float*)d_in[0];
  const int* ei   = (const int*)d_in[1];
  // d_in[2] = edge_type (unused by reference)
  const int* tei  = (const int*)d_in[3];
  const float* P[2][2][4];  // [stack][layer][W,a,g,b]
  int idx = 4;
  for (int st = 0; st < 2; ++st)
    for (int li = 0; li < 2; ++li)
      for (int p = 0; p < 4; ++p) P[st][li][p] = (const float*)d_in[idx++];
  const float* cW1 = (const float*)d_in[20];
  const float* cb1 = (const float*)d_in[21];
  const float* cW2 = (const float*)d_in[22];
  const float* cb2 = (const float*)d_in[23];

  float* ws     = (float*)d_ws;
  float* Wh     = ws;                                // N*64
  float* hpr    = Wh    + (size_t)N_NODES * HID;     // N*64
  float* cur    = hpr   + (size_t)N_NODES * HID;     // N*64
  float* spres  = cur   + (size_t)N_NODES * HID;     // N*64
  float* scores = spres + (size_t)N_NODES * HID;     // E
  float* scal   = scores + N_EDGES;                  // 4 scalars

  const int GB  = N_NODES / 16;                         // 6250 row stripes
  const int EB  = (N_EDGES + 255) / 256;
  const int SCB = (int)(((long long)N_EDGES * HID + 255) / 256);
  const int NB  = (N_NODES + 7) / 8;                    // 8 waves/block

  auto run_layer = [&](const float* in, int K,
                       const float* Wm, const float* a,
                       const float* g, const float* b,
                       const int* src, const int* dst, float* outbuf) {
    gemm_wmma_f32<<<GB, 128, 0, stream>>>(in, Wm, Wh, K);
    prep_scal<<<1, 2 * HID, 0, stream>>>(a, scal);
    hipMemsetAsync(hpr, 0, (size_t)N_NODES * HID * sizeof(float), stream);
    edge_scores_k<<<EB, 256, 0, stream>>>(Wh, src, dst, scal, scores, scal + 1);
    edge_expsum_k<<<EB, 256, 0, stream>>>(scores, scal, scal + 2);
    scatter_k<<<SCB, 256, 0, stream>>>(scores, scal, Wh, src, dst, hpr);
    elu_ln_k<<<NB, 256, 0, stream>>>(hpr, g, b, outbuf);
  };

  // spatial stack
  const int* ssrc = ei;  const int* sdst = ei + N_EDGES;
  run_layer(x,   IN_DIM, P[0][0][0], P[0][0][1], P[0][0][2], P[0][0][3], ssrc, sdst, cur);
  run_layer(cur, HID,    P[0][1][0], P[0][1][1], P[0][1][2], P[0][1][3], ssrc, sdst, spres);

  // temporal stack
  const int* tsrc = tei; const int* tdst = tei + N_EDGES;
  run_layer(x,   IN_DIM, P[1][0][0], P[1][0][1], P[1][0][2], P[1][0][3], tsrc, tdst, cur);
  run_layer(cur, HID,    P[1][1][0], P[1][1][1], P[1][1][2], P[1][1][3], tsrc, tdst, cur);

  // fused head: (sp + tp) -> relu fc -> sigmoid fc
  head_k<<<NB, 256, 0, stream>>>(spres, cur, cW1, cb1, cW2, cb2, (float*)d_out);
}